// RNN_11407433138271
// MI455X (gfx1250) — compile-verified
//
#include <hip/hip_runtime.h>
#include <hip/hip_bf16.h>

// Problem constants (match the JAX reference).
constexpr int NB = 64;    // batch
constexpr int TT = 512;   // timesteps
constexpr int DD = 512;   // input dim
constexpr int HH = 512;   // hidden dim
constexpr int NT = NB * TT;

constexpr int SCAN_BLOCKS = 32;   // one block per 16-col N-tile of the h state

typedef __attribute__((ext_vector_type(2))) float v2f;
typedef __attribute__((ext_vector_type(4))) float v4f;
typedef __attribute__((ext_vector_type(8))) float v8f;

// ---------------------------------------------------------------------------
// Phase 1: xw[i, h] = sum_d x[i, d] * Wx[d, h] + b[h]     (i = n*T + t)
// Each wave computes a 16(M) x 64(N) strip: A fragment loaded once per k-step,
// reused across 4 V_WMMA_F32_16X16X4_F32 issues.
// ---------------------------------------------------------------------------
__global__ void __launch_bounds__(256)
rnn_xproj_kernel(const float* __restrict__ x,    // (NT, D)
                 const float* __restrict__ Wx,   // (D, H)
                 const float* __restrict__ bias, // (H)
                 float* __restrict__ xw)         // (NT, H)
{
    const int lane = threadIdx.x & 31;
    const int wave = (blockIdx.x * blockDim.x + threadIdx.x) >> 5;

    constexpr int nStrips = HH / 64;           // 8
    const int mtile = wave / nStrips;          // 0..2047
    const int strip = wave % nStrips;          // 0..7
    const int m0 = mtile * 16;
    const int n0 = strip * 64;

    const int half = lane >> 4;                // 0: K={0,1}, 1: K={2,3}
    const int l    = lane & 15;

    v8f acc0 = {}, acc1 = {}, acc2 = {}, acc3 = {};

    const float* arow = x + (size_t)(m0 + l) * DD;   // one A row per lane
    #pragma unroll 4
    for (int k0 = 0; k0 < DD; k0 += 4) {
        const int ka = k0 + 2 * half;
        v2f a;
        a.x = arow[ka];
        a.y = arow[ka + 1];

        const float* brow = Wx + (size_t)ka * HH + n0 + l;
        v2f b0, b1, b2, b3;
        b0.x = brow[0];       b0.y = brow[HH];
        b1.x = brow[16];      b1.y = brow[HH + 16];
        b2.x = brow[32];      b2.y = brow[HH + 32];
        b3.x = brow[48];      b3.y = brow[HH + 48];

        acc0 = __builtin_amdgcn_wmma_f32_16x16x4_f32(false, a, false, b0, (short)0, acc0, false, false);
        acc1 = __builtin_amdgcn_wmma_f32_16x16x4_f32(false, a, false, b1, (short)0, acc1, false, false);
        acc2 = __builtin_amdgcn_wmma_f32_16x16x4_f32(false, a, false, b2, (short)0, acc2, false, false);
        acc3 = __builtin_amdgcn_wmma_f32_16x16x4_f32(false, a, false, b3, (short)0, acc3, false, false);
    }

    // D layout: VGPR j -> row M = m0 + j + 8*half, col N = n0 + 16*nn + l.
    v8f accs[4] = {acc0, acc1, acc2, acc3};
    #pragma unroll
    for (int nn = 0; nn < 4; ++nn) {
        const int col = n0 + nn * 16 + l;
        const float bv = bias[col];
        #pragma unroll
        for (int j = 0; j < 8; ++j) {
            const int row = m0 + j + 8 * half;
            xw[(size_t)row * HH + col] = accs[nn][j] + bv;
        }
    }
}

// ---------------------------------------------------------------------------
// Zero the grid-barrier counter (run once per call, before the scan).
// ---------------------------------------------------------------------------
__global__ void rnn_barrier_init_kernel(int* __restrict__ counter) {
    if (threadIdx.x == 0 && blockIdx.x == 0) *counter = 0;
}

// ---------------------------------------------------------------------------
// Phase 2: persistent scan kernel. All 512 timesteps in one launch.
//   h_t = tanh(xw_t + h_{t-1} @ Wh),  h state lives in `out` time slices.
//
// Block b owns output columns [16b, 16b+16); its 4 waves own the 4 M-tiles of
// the (64 x 512) state. The block's Wh column panel (512x16 fp32 = 32 KB) is
// staged into LDS ONCE, so the serial per-step K-chain reads B at LDS latency
// (pairs 64B apart -> ds_load_2addr) while A (h_{t-1}) streams from L2 via
// pipelined global b64 loads. Steps are separated by a device-scope
// monotonic-counter grid barrier (32 co-resident blocks).
// ---------------------------------------------------------------------------
__global__ void __launch_bounds__(128)
rnn_scan_kernel(const float* __restrict__ h0,   // (NB, H)
                const float* __restrict__ Wh,   // (H, H)
                const float* __restrict__ xw,   // (N, T, H)
                float* __restrict__ out,        // (N, T, H)
                int* __restrict__ barrier)
{
    __shared__ float sWh[HH * 16];              // [k][c] panel, 32 KB

    const int lane = threadIdx.x & 31;
    const int mtile = threadIdx.x >> 5;         // 0..3 (wave in block)
    const int m0 = mtile * 16;
    const int n0 = blockIdx.x * 16;             // block's column tile

    const int half = lane >> 4;
    const int l    = lane & 15;

    // ---- One-time: stage Wh[:, n0..n0+15] into LDS (float4 per thread). ----
    {
        const int rgrp = threadIdx.x >> 2;      // 0..31
        const int c4   = (threadIdx.x & 3) * 4; // 0,4,8,12
        #pragma unroll
        for (int k = 0; k < HH; k += 32) {
            const int kk = k + rgrp;
            const v4f v = *(const v4f*)(Wh + (size_t)kk * HH + n0 + c4);
            *(v4f*)(&sWh[kk * 16 + c4]) = v;
        }
    }
    __syncthreads();

    const int col = n0 + l;

    for (int t = 0; t < TT; ++t) {
        const float* hbase;
        size_t strideA;
        if (t == 0) { hbase = h0;                         strideA = HH; }
        else        { hbase = out + (size_t)(t - 1) * HH; strideA = (size_t)TT * HH; }
        const float* arow = hbase + (size_t)(m0 + l) * strideA;

        v8f acc = {};
        #pragma unroll 8
        for (int k0 = 0; k0 < HH; k0 += 4) {
            const int ka = k0 + 2 * half;
            v2f a;
            a.x = arow[ka];                    // global b64 (L2-resident)
            a.y = arow[ka + 1];
            v2f b;
            b.x = sWh[ka * 16 + l];            // ds pair, 64 B apart
            b.y = sWh[(ka + 1) * 16 + l];
            acc = __builtin_amdgcn_wmma_f32_16x16x4_f32(false, a, false, b, (short)0, acc, false, false);
        }

        // Fused +xw_t, tanh, and store of h_t into out[:, t, :].
        #pragma unroll
        for (int j = 0; j < 8; ++j) {
            const int n = m0 + j + 8 * half;   // batch row
            const size_t idx = ((size_t)n * TT + t) * HH + col;
            out[idx] = tanhf(acc[j] + xw[idx]);
        }

        // ---- Grid barrier: make h_t visible, then wait for all blocks. ----
        __threadfence();                        // release: stores -> device scope
        __syncthreads();
        if (threadIdx.x == 0) {
            __hip_atomic_fetch_add(barrier, 1, __ATOMIC_RELEASE, __HIP_MEMORY_SCOPE_AGENT);
            const int target = (t + 1) * SCAN_BLOCKS;
            while (__hip_atomic_load(barrier, __ATOMIC_ACQUIRE, __HIP_MEMORY_SCOPE_AGENT) < target) {
                __builtin_amdgcn_s_sleep(8);
            }
        }
        __syncthreads();
        __threadfence();                        // acquire: invalidate near caches
    }
}

// ---------------------------------------------------------------------------
extern "C" void kernel_launch(void* const* d_in, const int* in_sizes, int n_in,
                              void* d_out, int out_size, void* d_ws, size_t ws_size,
                              hipStream_t stream) {
    const float* x  = (const float*)d_in[0];   // (N, T, D)
    const float* h0 = (const float*)d_in[1];   // (N, H)
    const float* Wx = (const float*)d_in[2];   // (D, H)
    const float* Wh = (const float*)d_in[3];   // (H, H)
    const float* b  = (const float*)d_in[4];   // (H)
    float* out = (float*)d_out;                // (N, T, H)

    float* xw = (float*)d_ws;                  // (NT, H) = 64 MB scratch
    int* barrier = (int*)((char*)d_ws + (size_t)NT * HH * sizeof(float));

    // Reset the grid-barrier counter (ws is not re-initialized between replays).
    rnn_barrier_init_kernel<<<1, 32, 0, stream>>>(barrier);

    // Phase 1: big fp32 WMMA GEMM, 16384 waves (8 waves / 256-thread block).
    {
        const int totalWaves = (NT / 16) * (HH / 64);     // 16384
        const int blocks = totalWaves / 8;                // 2048
        rnn_xproj_kernel<<<blocks, 256, 0, stream>>>(x, Wx, b, xw);
    }

    // Phase 2: one persistent kernel scans all 512 timesteps.
    rnn_scan_kernel<<<SCAN_BLOCKS, 128, 0, stream>>>(h0, Wh, xw, out, barrier);
}